// DeformCeptionF2F_N_27049704030779
// MI455X (gfx1250) — compile-verified
//
#include <hip/hip_runtime.h>
#include <hip/hip_bf16.h>

// ---------------------------------------------------------------------------
// DeformCeption on MI455X (gfx1250).
// Heavy GEMMs (deform-conv einsum) run as fp32 WMMA (v_wmma_f32_16x16x4_f32):
// the workload is gather/memory bound (~7 GFLOP vs ~100MB traffic @23.3TB/s),
// so fp32 matrix math matches the reference precision at no roofline cost.
// Deformable sampling is materialized as an im2col buffer [M, K] in workspace
// so WMMA A fragments are contiguous float2 loads per lane; weights are
// repacked K-paired so B fragments are contiguous float2 loads as well.
// Each wave computes a 16x64 tile: 1 A load feeds 4 WMMAs per K step.
// Workspace requirement: ~121 MB.
// ---------------------------------------------------------------------------

typedef __attribute__((ext_vector_type(2))) float v2f;
typedef __attribute__((ext_vector_type(8))) float v8f;

#define EPS_BN 1e-5f

// ---------------- BN statistics: per-channel mean & inv-std ----------------
__global__ void dc_bn_stats(const float* __restrict__ x, float* __restrict__ mu,
                            float* __restrict__ istd, int C, int B, int HW) {
  int c = blockIdx.x;
  long n = (long)B * HW;
  float s = 0.f, s2 = 0.f;
  for (long i = threadIdx.x; i < n; i += blockDim.x) {
    long b = i / HW, p = i % HW;
    float v = x[(b * C + c) * (long)HW + p];
    s += v; s2 += v * v;
  }
  for (int o = 16; o > 0; o >>= 1) {
    s  += __shfl_down(s, o);
    s2 += __shfl_down(s2, o);
  }
  __shared__ float sh[16];
  int wid = threadIdx.x >> 5, lane = threadIdx.x & 31;
  if (lane == 0) { sh[wid] = s; sh[8 + wid] = s2; }
  __syncthreads();
  if (threadIdx.x == 0) {
    float ts = 0.f, ts2 = 0.f;
    int nw = blockDim.x >> 5;
    for (int i = 0; i < nw; ++i) { ts += sh[i]; ts2 += sh[8 + i]; }
    float m = ts / (float)n;
    float v = ts2 / (float)n - m * m;
    mu[c] = m;
    istd[c] = rsqrtf(fmaxf(v, 0.f) + EPS_BN);
  }
}

// ---------------- BN normalize + affine + ReLU -----------------------------
__global__ void dc_bn_relu(const float* __restrict__ x, const float* __restrict__ mu,
                           const float* __restrict__ istd, const float* __restrict__ g,
                           const float* __restrict__ bt, float* __restrict__ xn,
                           int C, int HW, long total) {
  long i = (long)blockIdx.x * blockDim.x + threadIdx.x;
  if (i >= total) return;
  int c = (int)((i / HW) % C);
  float v = (x[i] - mu[c]) * istd[c] * g[c] + bt[c];
  xn[i] = v > 0.f ? v : 0.f;
}

// ---------------- direct KxK conv (tiny Cout: offset convs) ----------------
__global__ void dc_direct_conv(const float* __restrict__ xn, const float* __restrict__ w,
                               const float* __restrict__ bias, float* __restrict__ out,
                               int B, int Cin, int Cout, int H, int W, int K) {
  int pad = (K - 1) / 2;
  long HW = (long)H * W;
  long total = (long)B * Cout * HW;
  long i = (long)blockIdx.x * blockDim.x + threadIdx.x;
  if (i >= total) return;
  int wq = (int)(i % W);
  int hq = (int)((i / W) % H);
  int o  = (int)((i / HW) % Cout);
  int b  = (int)(i / (HW * Cout));
  float acc = bias[o];
  for (int c = 0; c < Cin; ++c) {
    const float* xc = xn + ((long)(b * Cin + c)) * HW;
    const float* wc = w + (((long)o * Cin + c) * K) * K;
    for (int ky = 0; ky < K; ++ky) {
      int y = hq - pad + ky;
      if (y < 0 || y >= H) continue;
      for (int kx = 0; kx < K; ++kx) {
        int xq = wq - pad + kx;
        if (xq < 0 || xq >= W) continue;
        acc += xc[(long)y * W + xq] * wc[ky * K + kx];
      }
    }
  }
  out[i] = acc;
}

// ---------------- deformable im2col: col[m][c*KK+kk] -----------------------
// Bilinear taps with corner masking; coords computed once per (m,kk), reused
// across all C channels (channel stride HW).
__global__ void dc_deform_im2col(const float* __restrict__ xn, const float* __restrict__ off,
                                 float* __restrict__ col, int B, int C, int H, int W, int K) {
  int pad = (K - 1) / 2;
  int KK = K * K;
  long HW = (long)H * W;
  long M = (long)B * HW;
  long t = (long)blockIdx.x * blockDim.x + threadIdx.x;
  if (t >= M * KK) return;
  int kk = (int)(t % KK);
  long m = t / KK;
  int b = (int)(m / HW);
  int hw = (int)(m % HW);
  int h = hw / W, w = hw % W;
  int ky = kk / K, kx = kk % K;
  long obase = ((long)b * (2 * KK)) * HW + hw;
  float oy = off[obase + (long)(2 * kk + 0) * HW];
  float ox = off[obase + (long)(2 * kk + 1) * HW];
  float py = (float)(h - pad + ky) + oy;
  float px = (float)(w - pad + kx) + ox;
  float y0f = floorf(py), x0f = floorf(px);
  float dy = py - y0f, dx = px - x0f;
  int y0 = (int)y0f, x0 = (int)x0f;
  int y1 = y0 + 1, x1 = x0 + 1;
  float w00 = (1.f - dy) * (1.f - dx);
  float w01 = (1.f - dy) * dx;
  float w10 = dy * (1.f - dx);
  float w11 = dy * dx;
  bool vy0 = (y0 >= 0) && (y0 < H), vy1 = (y1 >= 0) && (y1 < H);
  bool vx0 = (x0 >= 0) && (x0 < W), vx1 = (x1 >= 0) && (x1 < W);
  w00 = (vy0 && vx0) ? w00 : 0.f;
  w01 = (vy0 && vx1) ? w01 : 0.f;
  w10 = (vy1 && vx0) ? w10 : 0.f;
  w11 = (vy1 && vx1) ? w11 : 0.f;
  int y0c = min(max(y0, 0), H - 1), y1c = min(max(y1, 0), H - 1);
  int x0c = min(max(x0, 0), W - 1), x1c = min(max(x1, 0), W - 1);
  int i00 = y0c * W + x0c, i01 = y0c * W + x1c;
  int i10 = y1c * W + x0c, i11 = y1c * W + x1c;
  const float* xb = xn + (long)b * C * HW;
  float* cp = col + m * (long)(C * KK) + kk;
  for (int c = 0; c < C; ++c) {
    const float* xc = xb + (long)c * HW;
    cp[(long)c * KK] = xc[i00] * w00 + xc[i01] * w01 + xc[i10] * w10 + xc[i11] * w11;
  }
}

// ---------------- small dense GEMM (O = 2 or 18, offset refinement) --------
__global__ void dc_small_gemm(const float* __restrict__ col, const float* __restrict__ w,
                              const float* __restrict__ bias, float* __restrict__ out,
                              long M, int Kt, int O, int HW) {
  long t = (long)blockIdx.x * blockDim.x + threadIdx.x;
  if (t >= M * O) return;
  int o = (int)(t % O);
  long m = t / O;
  const float* cr = col + m * (long)Kt;
  const float* wr = w + (long)o * Kt;
  float acc = bias[o];
  for (int k = 0; k < Kt; ++k) acc += cr[k] * wr[k];
  long b = m / HW, hw = m % HW;
  out[(b * O + o) * (long)HW + hw] = acc;
}

// ---------------- weight repack [O,Kt] -> K-paired [Kt/2][N][2] ------------
// wt2[((k>>1)*N + n)*2 + (k&1)] = w[n*Kt + k]  => B fragment = one float2 load
__global__ void dc_pack_w(const float* __restrict__ w, float* __restrict__ wt2,
                          int O, int Kt) {
  long i = (long)blockIdx.x * blockDim.x + threadIdx.x;
  if (i >= (long)O * Kt) return;
  int o = (int)(i / Kt), k = (int)(i % Kt);
  wt2[(((long)(k >> 1) * O + o) << 1) + (k & 1)] = w[i];
}

// ---------------- WMMA fp32 GEMM: out[m][n] = col[m][:] . W[n][:] + bias ---
// One wave computes a 16(M) x 64(N) tile with 4 v8f accumulators.
// Block = 4 waves stacked along M (64 rows); grid = (M/64, N/64).
// A 16x4 fp32 fragment: lane&15 = M row; lane>>4 selects K pair -> contiguous
//   float2 at col[row*Kt + k + 2*hi].
// B 4x16 fp32 fragment: lane&15 = N col; lane>>4 selects K pair -> contiguous
//   float2 at wt2[((k>>1)+hi)*N*2 + n*2].
// D: VGPR r -> M = mTile + r + 8*hi, N = nBase + t*16 + (lane&15).
__global__ void dc_wmma_gemm(const float* __restrict__ col, const float* __restrict__ wt2,
                             const float* __restrict__ bias, float* __restrict__ out,
                             int Kt, int N, int HW) {
  int wave = threadIdx.x >> 5;
  int lane = threadIdx.x & 31;
  int l15 = lane & 15;
  int hi = (lane >> 4) & 1;
  int mTile = blockIdx.x * 64 + wave * 16;
  int nBase = blockIdx.y * 64;

  const float* arow = col + (long)(mTile + l15) * Kt + 2 * hi;
  const float* brow = wt2 + ((long)hi * N + nBase + l15) * 2;
  const long bstep = (long)N * 2;   // advance one K pair

  v8f acc0 = {}, acc1 = {}, acc2 = {}, acc3 = {};
  for (int k = 0; k < Kt; k += 4) {
    v2f a = *(const v2f*)(arow + k);
    const float* bp = brow + (long)(k >> 1) * bstep;
    v2f b0 = *(const v2f*)(bp);
    v2f b1 = *(const v2f*)(bp + 32);
    v2f b2 = *(const v2f*)(bp + 64);
    v2f b3 = *(const v2f*)(bp + 96);
    __builtin_prefetch((const void*)(arow + k + 128), 0, 1);
    acc0 = __builtin_amdgcn_wmma_f32_16x16x4_f32(false, a, false, b0,
                                                 (short)0, acc0, false, false);
    acc1 = __builtin_amdgcn_wmma_f32_16x16x4_f32(false, a, false, b1,
                                                 (short)0, acc1, false, false);
    acc2 = __builtin_amdgcn_wmma_f32_16x16x4_f32(false, a, false, b2,
                                                 (short)0, acc2, false, false);
    acc3 = __builtin_amdgcn_wmma_f32_16x16x4_f32(false, a, false, b3,
                                                 (short)0, acc3, false, false);
  }

  long HWl = HW;
  v8f accs[4] = {acc0, acc1, acc2, acc3};
  for (int t = 0; t < 4; ++t) {
    int n = nBase + t * 16 + l15;
    float bv = bias[n];
    for (int r = 0; r < 8; ++r) {
      long m = (long)mTile + r + 8 * hi;
      long b_ = m / HWl, hw = m % HWl;
      out[(b_ * N + n) * HWl + hw] = accs[t][r] + bv;
    }
  }
}

// ---------------------------------------------------------------------------
extern "C" void kernel_launch(void* const* d_in, const int* in_sizes, int n_in,
                              void* d_out, int out_size, void* d_ws, size_t ws_size,
                              hipStream_t stream) {
  (void)in_sizes; (void)n_in; (void)out_size; (void)ws_size;
  const int B = 8, H = 32, W = 64, HW = H * W;
  const long M = (long)B * HW;          // 16384 spatial positions

  const float* x       = (const float*)d_in[0];
  const float* l0_bn_g = (const float*)d_in[1];
  const float* l0_bn_b = (const float*)d_in[2];
  const float* l0_off_w  = (const float*)d_in[3];
  const float* l0_off_b  = (const float*)d_in[4];
  const float* l0_offd_w = (const float*)d_in[5];
  const float* l0_offd_b = (const float*)d_in[6];
  const float* l0_dc_w   = (const float*)d_in[7];
  const float* l0_dc_b   = (const float*)d_in[8];
  const float* l1_bn_g = (const float*)d_in[9];
  const float* l1_bn_b = (const float*)d_in[10];
  const float* l1_off_w  = (const float*)d_in[11];
  const float* l1_off_b  = (const float*)d_in[12];
  const float* l1_offd_w = (const float*)d_in[13];
  const float* l1_offd_b = (const float*)d_in[14];
  const float* l1_dc_w   = (const float*)d_in[15];
  const float* l1_dc_b   = (const float*)d_in[16];
  float* out = (float*)d_out;

  // workspace carving (floats, 256B aligned chunks); total ~121 MB
  float* ws = (float*)d_ws;
  size_t p = 0;
  auto alloc = [&](size_t n) { float* r = ws + p; p += (n + 63) & ~(size_t)63; return r; };
  float* mu   = alloc(512);
  float* istd = alloc(512);
  float* xn   = alloc((size_t)B * 512 * HW);      // 32 MB (block0; reused by block1)
  float* hbuf = alloc((size_t)B * 128 * HW);      // 8 MB  (block0 output)
  float* coff = alloc((size_t)B * 18 * HW);       // offsets from plain conv
  float* doff = alloc((size_t)B * 18 * HW);       // offsets from deform conv
  float* wt   = alloc((size_t)1152 * 128);        // K-paired main weights
  float* colb = alloc((size_t)M * 1152);          // 72 MB im2col buffer (reused)

  const int TB = 256;
  auto blocks = [](long n, int tb) { return (unsigned)((n + tb - 1) / tb); };

  // ============================ Block 0 (K=1) ==============================
  {
    const int C = 512, K = 1, KK = 1, Kt = C * KK, O = 128, OFFC = 2;
    long tot = (long)B * C * HW;
    dc_bn_stats<<<C, TB, 0, stream>>>(x, mu, istd, C, B, HW);
    dc_bn_relu<<<blocks(tot, TB), TB, 0, stream>>>(x, mu, istd, l0_bn_g, l0_bn_b,
                                                   xn, C, HW, tot);
    long toff = (long)B * OFFC * HW;
    dc_direct_conv<<<blocks(toff, TB), TB, 0, stream>>>(xn, l0_off_w, l0_off_b, coff,
                                                        B, C, OFFC, H, W, K);
    dc_deform_im2col<<<blocks(M * KK, TB), TB, 0, stream>>>(xn, coff, colb, B, C, H, W, K);
    dc_small_gemm<<<blocks(M * OFFC, TB), TB, 0, stream>>>(colb, l0_offd_w, l0_offd_b,
                                                           doff, M, Kt, OFFC, HW);
    dc_deform_im2col<<<blocks(M * KK, TB), TB, 0, stream>>>(xn, doff, colb, B, C, H, W, K);
    dc_pack_w<<<blocks((long)O * Kt, TB), TB, 0, stream>>>(l0_dc_w, wt, O, Kt);
    dim3 grid((unsigned)(M / 64), (unsigned)(O / 64));
    dc_wmma_gemm<<<grid, 128, 0, stream>>>(colb, wt, l0_dc_b, hbuf, Kt, O, HW);
  }

  // ============================ Block 1 (K=3) ==============================
  {
    const int C = 128, K = 3, KK = 9, Kt = C * KK, O = 128, OFFC = 18;
    long tot = (long)B * C * HW;
    dc_bn_stats<<<C, TB, 0, stream>>>(hbuf, mu, istd, C, B, HW);
    dc_bn_relu<<<blocks(tot, TB), TB, 0, stream>>>(hbuf, mu, istd, l1_bn_g, l1_bn_b,
                                                   xn, C, HW, tot);
    long toff = (long)B * OFFC * HW;
    dc_direct_conv<<<blocks(toff, TB), TB, 0, stream>>>(xn, l1_off_w, l1_off_b, coff,
                                                        B, C, OFFC, H, W, K);
    dc_deform_im2col<<<blocks(M * KK, TB), TB, 0, stream>>>(xn, coff, colb, B, C, H, W, K);
    dc_small_gemm<<<blocks(M * OFFC, TB), TB, 0, stream>>>(colb, l1_offd_w, l1_offd_b,
                                                           doff, M, Kt, OFFC, HW);
    dc_deform_im2col<<<blocks(M * KK, TB), TB, 0, stream>>>(xn, doff, colb, B, C, H, W, K);
    dc_pack_w<<<blocks((long)O * Kt, TB), TB, 0, stream>>>(l1_dc_w, wt, O, Kt);
    dim3 grid((unsigned)(M / 64), (unsigned)(O / 64));
    dc_wmma_gemm<<<grid, 128, 0, stream>>>(colb, wt, l1_dc_b, out, Kt, O, HW);
  }
}